// _MultiHeadAttentionConv_11158325035084
// MI455X (gfx1250) — compile-verified
//
#include <hip/hip_runtime.h>

#define D_IN   128
#define D_OUTC 128
#define HHEADS 4

typedef float v2f __attribute__((ext_vector_type(2)));
typedef float v8f __attribute__((ext_vector_type(8)));

// ---- order-preserving float<->int encoding for integer atomicMax ----
__device__ __forceinline__ int ord_of_float(float f) {
    int i = __float_as_int(f);
    return (i >= 0) ? i : (i ^ 0x7FFFFFFF);
}
__device__ __forceinline__ float float_of_ord(int k) {
    int i = (k >= 0) ? k : (k ^ 0x7FFFFFFF);
    return __int_as_float(i);
}
__device__ __forceinline__ float elu1(float x) {
    return (x > 0.0f) ? x : (__expf(x) - 1.0f);
}

// ---------------------------------------------------------------------------
// K1: H = X @ W   (nrows x 128) @ (128 x 128), FP32 WMMA 16x16x4.
// One 256-thread block (8 wave32) per 16-row block; wave w owns cols [16w,16w+16).
// A tile staged in LDS; B fragments read from global (W is 64KB, L2/L0 resident).
// ---------------------------------------------------------------------------
__global__ void __launch_bounds__(256)
gat_gemm_wmma(const float* __restrict__ X, const float* __restrict__ W,
              float* __restrict__ Hout, int nrows) {
    __shared__ float As[16 * 132];                 // +4 pad per row: conflict-free
    const int node_base = blockIdx.x * 16;
    const int t    = threadIdx.x;
    const int lane = t & 31;
    const int wave = t >> 5;                       // 0..7

    // cooperative stage of the 16x128 A tile (2048 floats / 256 threads)
    #pragma unroll
    for (int i = 0; i < 8; ++i) {
        int idx = t + i * 256;
        int r = idx >> 7, c = idx & 127;
        int rr = node_base + r;
        rr = (rr < nrows) ? rr : (nrows - 1);
        As[r * 132 + c] = X[(size_t)rr * D_IN + c];
    }
    __syncthreads();

    const int half = lane >> 4;                    // K sub-slice select
    const int l15  = lane & 15;
    const int col_base = wave * 16;

    v8f c = {};
    #pragma unroll
    for (int k0 = 0; k0 < 128; k0 += 4) {
        const int kb = k0 + half * 2;
        v2f a, b;
        a.x = As[l15 * 132 + kb];
        a.y = As[l15 * 132 + kb + 1];
        b.x = W[(size_t)kb       * D_OUTC + col_base + l15];
        b.y = W[(size_t)(kb + 1) * D_OUTC + col_base + l15];
        // D = A(16x4) * B(4x16) + C  -> v_wmma_f32_16x16x4_f32
        c = __builtin_amdgcn_wmma_f32_16x16x4_f32(false, a, false, b,
                                                  (short)0, c, false, false);
    }

    #pragma unroll
    for (int v = 0; v < 8; ++v) {
        int row = node_base + v + 8 * half;
        if (row < nrows)
            Hout[(size_t)row * D_OUTC + col_base + l15] = c[v];
    }
}

// ---------------------------------------------------------------------------
// K2: per-node head scores  s[n,h] = <H[n, 32h:32h+32], a[off:off+32]>
// One wave per node; lane l holds dims 4l..4l+3 (head = l>>3), xor-shuffle reduce.
// ---------------------------------------------------------------------------
__global__ void __launch_bounds__(256)
gat_scores(const float* __restrict__ Hin, const float* __restrict__ a_vec,
           float* __restrict__ S, int nnodes, int a_off) {
    const int node = (int)((blockIdx.x * (size_t)blockDim.x + threadIdx.x) >> 5);
    const int lane = threadIdx.x & 31;
    if (node >= nnodes) return;
    const float4 hv = *(const float4*)(Hin + (size_t)node * D_OUTC + lane * 4);
    const int head = lane >> 3;
    const int off  = a_off + (lane & 7) * 4;
    float p = hv.x * a_vec[off] + hv.y * a_vec[off + 1] +
              hv.z * a_vec[off + 2] + hv.w * a_vec[off + 3];
    p += __shfl_xor(p, 1, 32);
    p += __shfl_xor(p, 2, 32);
    p += __shfl_xor(p, 4, 32);
    if ((lane & 7) == 0) S[(size_t)node * HHEADS + head] = p;
}

// ---------------------------------------------------------------------------
// K_init: zero d_out & denoms, seed segment-max with -inf encoding.
// ---------------------------------------------------------------------------
__global__ void gat_init(float* __restrict__ out, size_t n_out,
                         float* __restrict__ denom, int* __restrict__ mord, int n_nd) {
    size_t i = (size_t)blockIdx.x * blockDim.x + threadIdx.x;
    if (i < n_out) out[i] = 0.0f;
    if (i < (size_t)n_nd) { denom[i] = 0.0f; mord[i] = (int)0x80000000; }
}

// ---------------------------------------------------------------------------
// K3: segment max of edge logits (ordered-int atomicMax).
// ---------------------------------------------------------------------------
__global__ void __launch_bounds__(256)
gat_edge_max(const int* __restrict__ src_idx, const int* __restrict__ dst_idx,
             const float* __restrict__ Ssrc, const float* __restrict__ Sdst,
             int* __restrict__ mord, int E) {
    int e = blockIdx.x * blockDim.x + threadIdx.x;
    if (e >= E) return;
    const int si = src_idx[e], di = dst_idx[e];
    #pragma unroll
    for (int h = 0; h < HHEADS; ++h) {
        float ev = elu1(Ssrc[si * HHEADS + h] + Sdst[di * HHEADS + h]);
        atomicMax(&mord[di * HHEADS + h], ord_of_float(ev));
    }
}

// ---------------------------------------------------------------------------
// K4: segment sum of exp(e - max).
// ---------------------------------------------------------------------------
__global__ void __launch_bounds__(256)
gat_edge_denom(const int* __restrict__ src_idx, const int* __restrict__ dst_idx,
               const float* __restrict__ Ssrc, const float* __restrict__ Sdst,
               const int* __restrict__ mord, float* __restrict__ denom, int E) {
    int e = blockIdx.x * blockDim.x + threadIdx.x;
    if (e >= E) return;
    const int si = src_idx[e], di = dst_idx[e];
    #pragma unroll
    for (int h = 0; h < HHEADS; ++h) {
        float ev = elu1(Ssrc[si * HHEADS + h] + Sdst[di * HHEADS + h]);
        float m  = float_of_ord(mord[di * HHEADS + h]);
        atomicAdd(&denom[di * HHEADS + h], __expf(ev - m));
    }
}

// ---------------------------------------------------------------------------
// K5: out[dst] += alpha * h_src[src]  (wave per edge; float4/lane; f32 atomics;
// both h_src gather and out scatter are L2-resident on the 192MB L2).
// ---------------------------------------------------------------------------
__global__ void __launch_bounds__(256)
gat_aggregate(const int* __restrict__ src_idx, const int* __restrict__ dst_idx,
              const float* __restrict__ Ssrc, const float* __restrict__ Sdst,
              const int* __restrict__ mord, const float* __restrict__ denom,
              const float* __restrict__ Hsrc, float* __restrict__ out, int E) {
    const int e    = (int)((blockIdx.x * (size_t)blockDim.x + threadIdx.x) >> 5);
    const int lane = threadIdx.x & 31;
    if (e >= E) return;
    const int si = src_idx[e], di = dst_idx[e];
    const int head = lane >> 3;
    float ev  = elu1(Ssrc[si * HHEADS + head] + Sdst[di * HHEADS + head]);
    float m   = float_of_ord(mord[di * HHEADS + head]);
    float den = denom[di * HHEADS + head];
    float alpha = __expf(ev - m) / (den + 1e-12f);
    const float4 hv = *(const float4*)(Hsrc + (size_t)si * D_OUTC + lane * 4);
    float* o = out + (size_t)di * D_OUTC + lane * 4;
    atomicAdd(o + 0, alpha * hv.x);
    atomicAdd(o + 1, alpha * hv.y);
    atomicAdd(o + 2, alpha * hv.z);
    atomicAdd(o + 3, alpha * hv.w);
}

// ---------------------------------------------------------------------------
extern "C" void kernel_launch(void* const* d_in, const int* in_sizes, int n_in,
                              void* d_out, int out_size, void* d_ws, size_t ws_size,
                              hipStream_t stream) {
    const float* src_feat = (const float*)d_in[0];
    const float* dst_feat = (const float*)d_in[1];
    const int*   edge     = (const int*)d_in[2];
    const float* W        = (const float*)d_in[3];
    const float* a_vec    = (const float*)d_in[4];
    float* out = (float*)d_out;

    const int n_src = in_sizes[0] / D_IN;
    const int n_dst = in_sizes[1] / D_IN;
    const int E     = in_sizes[2] / 2;
    const int* src_idx = edge;
    const int* dst_idx = edge + E;

    // workspace layout (floats)
    float* ws = (float*)d_ws;
    size_t off = 0;
    float* h_src = ws + off; off += (size_t)n_src * D_OUTC;
    float* h_dst = ws + off; off += (size_t)n_dst * D_OUTC;
    float* s_src = ws + off; off += (size_t)n_src * HHEADS;
    float* s_dst = ws + off; off += (size_t)n_dst * HHEADS;
    int*   mord  = (int*)(ws + off); off += (size_t)n_dst * HHEADS;
    float* denom = ws + off; off += (size_t)n_dst * HHEADS;

    // init output / segment buffers (buffers are poisoned between runs)
    const size_t n_out_f = (size_t)n_dst * D_OUTC;
    const int    n_nd    = n_dst * HHEADS;
    {
        size_t n = n_out_f > (size_t)n_nd ? n_out_f : (size_t)n_nd;
        gat_init<<<(unsigned)((n + 255) / 256), 256, 0, stream>>>(out, n_out_f, denom, mord, n_nd);
    }

    // node GEMMs (FP32 WMMA)
    gat_gemm_wmma<<<(n_src + 15) / 16, 256, 0, stream>>>(src_feat, W, h_src, n_src);
    gat_gemm_wmma<<<(n_dst + 15) / 16, 256, 0, stream>>>(dst_feat, W, h_dst, n_dst);

    // per-node head scores (a_src = a[0:32], a_dst = a[32:64])
    gat_scores<<<(n_src + 7) / 8, 256, 0, stream>>>(h_src, a_vec, s_src, n_src, 0);
    gat_scores<<<(n_dst + 7) / 8, 256, 0, stream>>>(h_dst, a_vec, s_dst, n_dst, 32);

    // edge softmax (segment max, then denom)
    gat_edge_max  <<<(E + 255) / 256, 256, 0, stream>>>(src_idx, dst_idx, s_src, s_dst, mord, E);
    gat_edge_denom<<<(E + 255) / 256, 256, 0, stream>>>(src_idx, dst_idx, s_src, s_dst, mord, denom, E);

    // weighted aggregation
    gat_aggregate<<<(E + 7) / 8, 256, 0, stream>>>(src_idx, dst_idx, s_src, s_dst,
                                                   mord, denom, h_src, out, E);
}